// TopKSAE_65206193488019
// MI455X (gfx1250) — compile-verified
//
#include <hip/hip_runtime.h>

typedef float v2f __attribute__((ext_vector_type(2)));
typedef float v4f __attribute__((ext_vector_type(4)));
typedef float v8f __attribute__((ext_vector_type(8)));

#define B_ROWS 4096
#define D_DIM  1024
#define N_DIM  16384
#define K_TOP  32

#define BM 128
#define BN 128
#define BK 32
#define LDT (BK + 1)   // padded LDS row stride (33 floats) -> conflict-free column reads

// -------------------------------------------------------------------------
// Kernel 1: h_pre = (x - b_dec) @ W_enc^T + b_enc   via V_WMMA_F32_16X16X4_F32
// grid = (N/BN, B/BM), 256 threads (8 wave32)
// -------------------------------------------------------------------------
__global__ __launch_bounds__(256) void sae_encode_wmma(
    const float* __restrict__ x,
    const float* __restrict__ W_enc,
    const float* __restrict__ b_enc,
    const float* __restrict__ b_dec,
    float* __restrict__ h_pre)
{
    __shared__ float As[BM * LDT];   // (x - b_dec) tile, [m][k]
    __shared__ float Bs[BN * LDT];   // W_enc tile,      [n][k]

    const int tid   = threadIdx.x;
    const int lane  = tid & 31;
    const int wave  = tid >> 5;
    const int l16   = lane & 15;
    const int lhalf = lane >> 4;     // 0: K=k..k+1 / M-rows 0..7 ; 1: K=k+2..k+3 / M-rows 8..15

    const int n0 = blockIdx.x * BN;
    const int m0 = blockIdx.y * BM;

    const int wm = wave & 1;         // 64-row half of the M tile
    const int wn = wave >> 1;        // 32-col slice of the N tile

    v8f acc[4][2];
    {
        v8f zero = {0.f, 0.f, 0.f, 0.f, 0.f, 0.f, 0.f, 0.f};
        for (int tm = 0; tm < 4; ++tm)
            for (int tn = 0; tn < 2; ++tn)
                acc[tm][tn] = zero;
    }

    for (int k0 = 0; k0 < D_DIM; k0 += BK) {
        // Stage A = x - b_dec (128 rows x 32 k): 1024 float4 loads / 256 threads
        for (int i = 0; i < 4; ++i) {
            int linear = tid + i * 256;          // 0..1023
            int r  = linear >> 3;                // row within tile
            int c4 = (linear & 7) << 2;          // float4 column
            v4f xv = *(const v4f*)(x + (size_t)(m0 + r) * D_DIM + k0 + c4);
            v4f bd = *(const v4f*)(b_dec + k0 + c4);
            float* dst = As + r * LDT + c4;
            dst[0] = xv.x - bd.x; dst[1] = xv.y - bd.y;
            dst[2] = xv.z - bd.z; dst[3] = xv.w - bd.w;
        }
        // Stage B = W_enc (128 n-rows x 32 k)
        for (int i = 0; i < 4; ++i) {
            int linear = tid + i * 256;
            int r  = linear >> 3;
            int c4 = (linear & 7) << 2;
            v4f wv = *(const v4f*)(W_enc + (size_t)(n0 + r) * D_DIM + k0 + c4);
            float* dst = Bs + r * LDT + c4;
            dst[0] = wv.x; dst[1] = wv.y; dst[2] = wv.z; dst[3] = wv.w;
        }
        __syncthreads();

        for (int kk = 0; kk < BK; kk += 4) {
            const int ka = kk + 2 * lhalf;       // per-lane K offset (ISA 32-bit A/B layout)
            v2f afrag[4], bfrag[2];
            for (int tm = 0; tm < 4; ++tm) {
                const float* ap = As + (wm * 64 + tm * 16 + l16) * LDT + ka;
                afrag[tm].x = ap[0];
                afrag[tm].y = ap[1];
            }
            for (int tn = 0; tn < 2; ++tn) {
                const float* bp = Bs + (wn * 32 + tn * 16 + l16) * LDT + ka;
                bfrag[tn].x = bp[0];
                bfrag[tn].y = bp[1];
            }
            for (int tm = 0; tm < 4; ++tm)
                for (int tn = 0; tn < 2; ++tn)
                    acc[tm][tn] = __builtin_amdgcn_wmma_f32_16x16x4_f32(
                        false, afrag[tm], false, bfrag[tn],
                        (short)0, acc[tm][tn], false, false);
        }
        __syncthreads();
    }

    // Write h_pre (+ b_enc). C/D layout: VGPR r -> M = r (+8 for lanes 16-31), N = lane%16.
    for (int tn = 0; tn < 2; ++tn) {
        const int col = n0 + wn * 32 + tn * 16 + l16;
        const float be = b_enc[col];
        for (int tm = 0; tm < 4; ++tm) {
            const int rowbase = m0 + wm * 64 + tm * 16 + lhalf * 8;
            for (int r = 0; r < 8; ++r) {
                h_pre[(size_t)(rowbase + r) * N_DIM + col] = acc[tm][tn][r] + be;
            }
        }
    }
}

// -------------------------------------------------------------------------
// Kernel 2: per-row top-32 -> h_sparse, then sparse decode x_hat.
// One block (256 threads) per row; h_pre row lives in dynamic LDS.
// -------------------------------------------------------------------------
__global__ __launch_bounds__(256) void sae_topk_decode(
    const float* __restrict__ h_pre,
    const float* __restrict__ W_dec,
    const float* __restrict__ b_dec,
    float* __restrict__ h_sparse,
    float* __restrict__ x_hat)
{
    extern __shared__ float smem[];
    float* vals = smem;                       // N_DIM floats
    float* redv = vals + N_DIM;               // 256
    int*   redi = (int*)(redv + 256);         // 256
    float* topv = (float*)(redi + 256);       // K_TOP
    int*   topi = (int*)(topv + K_TOP);       // K_TOP

    const int tid = threadIdx.x;
    const int row = blockIdx.x;
    const float* src = h_pre + (size_t)row * N_DIM;

    for (int i = tid; i < N_DIM / 4; i += 256) {
        *(v4f*)(vals + i * 4) = *(const v4f*)(src + i * 4);
    }
    __syncthreads();

    const unsigned NEG_INF_BITS = 0xff800000u;
    const float NEG_INF = __int_as_float((int)NEG_INF_BITS);

    for (int it = 0; it < K_TOP; ++it) {
        // strided scan: lane-consecutive LDS addresses -> conflict-free
        float bv = NEG_INF;
        int bi = 0;
        for (int j = 0; j < N_DIM / 256; ++j) {
            const int idx = j * 256 + tid;
            const float v = vals[idx];
            if (v > bv) { bv = v; bi = idx; }
        }
        redv[tid] = bv;
        redi[tid] = bi;
        __syncthreads();
        for (int s = 128; s > 0; s >>= 1) {
            if (tid < s) {
                if (redv[tid + s] > redv[tid]) {
                    redv[tid] = redv[tid + s];
                    redi[tid] = redi[tid + s];
                }
            }
            __syncthreads();
        }
        if (tid == 0) {
            topv[it] = redv[0];
            topi[it] = redi[0];
            vals[redi[0]] = NEG_INF;           // sentinel marks "selected"
        }
        __syncthreads();
    }

    // h_sparse row: 0 everywhere, original value at selected indices
    float* hs = h_sparse + (size_t)row * N_DIM;
    for (int i = tid; i < N_DIM / 4; i += 256) {
        v4f v = *(const v4f*)(vals + i * 4);
        v4f o;
        for (int c = 0; c < 4; ++c) {
            float outv = 0.0f;
            if (__float_as_uint(v[c]) == NEG_INF_BITS) {
                const int gi = i * 4 + c;
                for (int t = 0; t < K_TOP; ++t)
                    if (topi[t] == gi) outv = topv[t];
            }
            o[c] = outv;
        }
        *(v4f*)(hs + i * 4) = o;
    }

    // decode: x_hat[row][d] = b_dec[d] + sum_t topv[t] * W_dec[d][topi[t]]
    // (W_dec is 64 MB -> L2-resident, scattered reads served from L2)
    float* xh = x_hat + (size_t)row * D_DIM;
    for (int d = tid; d < D_DIM; d += 256) {
        float acc = b_dec[d];
        for (int t = 0; t < K_TOP; ++t) {
            acc += topv[t] * W_dec[(size_t)d * N_DIM + topi[t]];
        }
        xh[d] = acc;
    }
}

// -------------------------------------------------------------------------
extern "C" void kernel_launch(void* const* d_in, const int* in_sizes, int n_in,
                              void* d_out, int out_size, void* d_ws, size_t ws_size,
                              hipStream_t stream) {
    const float* x     = (const float*)d_in[0];
    const float* W_enc = (const float*)d_in[1];
    const float* b_enc = (const float*)d_in[2];
    const float* W_dec = (const float*)d_in[3];
    const float* b_dec = (const float*)d_in[4];
    // d_in[5] is k (single int on device); fixed problem constant K_TOP = 32.

    float* x_hat    = (float*)d_out;
    float* h_sparse = x_hat + (size_t)B_ROWS * D_DIM;
    float* h_pre    = h_sparse + (size_t)B_ROWS * N_DIM;

    dim3 gridA(N_DIM / BN, B_ROWS / BM);
    sae_encode_wmma<<<gridA, 256, 0, stream>>>(x, W_enc, b_enc, b_dec, h_pre);

    const size_t smemB = (size_t)(N_DIM + 256) * sizeof(float)
                       + 256 * sizeof(int)
                       + K_TOP * sizeof(float)
                       + K_TOP * sizeof(int);
    sae_topk_decode<<<B_ROWS, 256, smemB, stream>>>(h_pre, W_dec, b_dec, h_sparse, x_hat);
}